// LabelPropagation_37211596652632
// MI455X (gfx1250) — compile-verified
//
#include <hip/hip_runtime.h>
#include <cstdint>
#include <cstddef>

// ---------------------------------------------------------------------------
// Label propagation on MI455X (gfx1250, wave32).
// Roofline: ~8.7 GB of L2-resident traffic over 10 layers; the working set
// (~105 MB: 3 node-feature buffers (in, out, y) + CSR edge pairs) fits in the 192 MB L2.
// Strategy: build CSR once per call so the 10 hot layers are pure gathers
// (no fp atomics); stage each node's edge stream with gfx1250 async
// global->LDS copies (ASYNCcnt), two chunks in flight, one wave per node
// (lane owns 2 classes as a float2 => one coalesced 256B row gather/edge).
// WMMA is deliberately unused: dense reformulation would be 1.28 TFLOP/layer
// vs ~0.41 GFLOP/layer sparse.
// ---------------------------------------------------------------------------

#define NN 100000          // nodes  (NN % 4 == 0)
#define NE 3200000         // edges
#define NC 64              // classes
#define NLAYERS 10
#define ALPHA 0.9f

// ------------------------------ helpers -----------------------------------

__device__ __forceinline__ void async_copy_pair_to_lds(const int2* gsrc, void* lds_dst) {
    // gfx1250: GLOBAL_LOAD_ASYNC_TO_LDS_B64 (tracked by ASYNCcnt).
    // vdst VGPR = LDS byte offset (flat shared addr truncates to LDS offset),
    // vaddr = 64-bit global address, saddr = off.
    unsigned int lds_off = (unsigned int)(uintptr_t)lds_dst;
    asm volatile("global_load_async_to_lds_b64 %0, %1, off"
                 :: "v"(lds_off), "v"(gsrc)
                 : "memory");
}
__device__ __forceinline__ void wait_async0() {
    asm volatile("s_wait_asynccnt 0" ::: "memory");
}
__device__ __forceinline__ void wait_async1() {
    // in-order completion => oldest outstanding chunk is done
    asm volatile("s_wait_asynccnt 1" ::: "memory");
}

// ------------------------------ kernels ------------------------------------

__global__ void k_zero_i32(int* __restrict__ p, int n) {
    int i = blockIdx.x * blockDim.x + threadIdx.x;
    if (i < n) p[i] = 0;
}

__global__ void k_count_deg(const int* __restrict__ dst, int* __restrict__ deg) {
    int e = blockIdx.x * blockDim.x + threadIdx.x;
    if (e < NE) {
        int pf = e + 16384; if (pf >= NE) pf = NE - 1;
        __builtin_prefetch(&dst[pf], 0, 1);            // global_prefetch_b8
        atomicAdd(&deg[dst[e]], 1);
    }
}

// Single-block exclusive scan of deg[NN] -> row_ptr[NN+1].
// 1024 threads, 4 elements/thread/tile, shfl wave scans: 3 barriers per
// 4096-element tile (25 tiles) instead of ~20 barriers per 1024 elements.
__global__ void k_scan(const int* __restrict__ deg, int* __restrict__ row_ptr) {
    __shared__ int wsum[32];
    const int tid  = threadIdx.x;
    const int lane = tid & 31;
    const int w    = tid >> 5;
    const int4* deg4 = (const int4*)deg;
    int carry = 0;
    for (int base = 0; base < NN; base += 4096) {
        const int i0 = base + tid * 4;
        int4 a = make_int4(0, 0, 0, 0);
        if (i0 < NN) a = deg4[i0 >> 2];                // NN % 4 == 0 => safe
        const int tsum = a.x + a.y + a.z + a.w;
        // wave-inclusive scan of per-thread sums
        int s = tsum;
        #pragma unroll
        for (int off = 1; off < 32; off <<= 1) {
            int t = __shfl_up(s, off, 32);
            if (lane >= off) s += t;
        }
        if (lane == 31) wsum[w] = s;
        __syncthreads();
        if (w == 0) {
            int x = wsum[lane];
            #pragma unroll
            for (int off = 1; off < 32; off <<= 1) {
                int t = __shfl_up(x, off, 32);
                if (lane >= off) x += t;
            }
            wsum[lane] = x;                            // inclusive wave totals
        }
        __syncthreads();
        const int wave_off = (w > 0) ? wsum[w - 1] : 0;
        if (i0 < NN) {
            int excl = carry + wave_off + (s - tsum);  // exclusive @ first elem
            row_ptr[i0 + 0] = excl; excl += a.x;
            row_ptr[i0 + 1] = excl; excl += a.y;
            row_ptr[i0 + 2] = excl; excl += a.z;
            row_ptr[i0 + 3] = excl;
        }
        carry += wsum[31];
        __syncthreads();                               // protect wsum reuse
    }
    if (tid == 0) row_ptr[NN] = carry;
}

__global__ void k_norm_cursor(const int* __restrict__ deg, const int* __restrict__ row_ptr,
                              float* __restrict__ norm, int* __restrict__ cursor) {
    int i = blockIdx.x * blockDim.x + threadIdx.x;
    if (i < NN) {
        norm[i]   = rsqrtf(fmaxf((float)deg[i], 1.0f));
        cursor[i] = row_ptr[i];
    }
}

// CSR scatter; folds norm[src] into the stored edge weight so the 10 hot
// layers never touch norm[src]. pairs[pos] = (src, ew*norm[src]).
__global__ void k_scatter(const int* __restrict__ src, const int* __restrict__ dst,
                          const float* __restrict__ ew, const float* __restrict__ norm,
                          int* __restrict__ cursor, int2* __restrict__ pairs) {
    int e = blockIdx.x * blockDim.x + threadIdx.x;
    if (e < NE) {
        int pf = e + 16384; if (pf >= NE) pf = NE - 1;
        __builtin_prefetch(&src[pf], 0, 1);
        __builtin_prefetch(&ew[pf], 0, 1);
        int d = dst[e];
        int s = src[e];
        float wgt = ew[e] * norm[s];
        int pos = atomicAdd(&cursor[d], 1);
        int2 p; p.x = s; p.y = __float_as_int(wgt);
        pairs[pos] = p;
    }
}

// out = mask ? y : 0, vectorized float4 (NC % 4 == 0).
__global__ void k_init(const float* __restrict__ y, const int* __restrict__ mask,
                       float* __restrict__ out) {
    int i = blockIdx.x * blockDim.x + threadIdx.x;    // float4 index
    if (i < (NN * NC) / 4) {
        const int node = (i * 4) >> 6;
        float4 v = ((const float4*)y)[i];
        if (mask[node] == 0) v = make_float4(0.f, 0.f, 0.f, 0.f);
        ((float4*)out)[i] = v;
    }
}

// One wave per node. Lane owns classes (2*lane, 2*lane+1) as a float2.
// Edge stream staged via async global->LDS b64, TWO 32-edge chunks in
// flight; chunk k lives in buffer k&1 and is re-issued for chunk k+2
// right after being consumed.
__launch_bounds__(256)
__global__ void k_layer(const float* __restrict__ in, float* __restrict__ out,
                        const int2* __restrict__ pairs, const int* __restrict__ row_ptr,
                        const float* __restrict__ norm, const float* __restrict__ y,
                        const int* __restrict__ mask) {
    const int lane = threadIdx.x & 31;
    const int wl   = threadIdx.x >> 5;                 // wave within block (8)
    const int node = blockIdx.x * (blockDim.x >> 5) + wl;
    __shared__ int2 stage[8][2][32];                   // 4 KB per block
    if (node >= NN) return;                            // no cross-wave sync below

    const int beg = row_ptr[node];
    const int end = row_ptr[node + 1];
    const int nchunks = (end - beg + 31) >> 5;
    float ax = 0.0f, ay = 0.0f;

    // prologue: two chunks in flight
    if (nchunks > 0) {
        int e = beg + lane;
        if (e < end) async_copy_pair_to_lds(&pairs[e], &stage[wl][0][lane]);
    }
    if (nchunks > 1) {
        int e = beg + 32 + lane;
        if (e < end) async_copy_pair_to_lds(&pairs[e], &stage[wl][1][lane]);
    }

    for (int ci = 0; ci < nchunks; ++ci) {
        if (ci + 1 < nchunks) wait_async1();           // chunk ci complete
        else                  wait_async0();
        const int cb  = beg + (ci << 5);
        const int buf = ci & 1;
        int cnt = end - cb; if (cnt > 32) cnt = 32;
        #pragma unroll 4
        for (int j = 0; j < cnt; ++j) {
            int2  p = stage[wl][buf][j];               // ds_load_b64 broadcast
            float wv = __int_as_float(p.y);
            const float2 v =
                ((const float2*)(in + ((size_t)(unsigned)p.x << 6)))[lane];
            ax = fmaf(wv, v.x, ax);
            ay = fmaf(wv, v.y, ay);
        }
        if (ci + 2 < nchunks) {                        // refill consumed buffer
            int e = cb + 64 + lane;
            if (e < end) async_copy_pair_to_lds(&pairs[e], &stage[wl][buf][lane]);
        }
    }

    const float nn = norm[node];
    const int   m  = mask[node];
    const float2 yv = ((const float2*)(y + ((size_t)node << 6)))[lane];
    float lx = m ? (1.0f - ALPHA) * yv.x : 0.0f;
    float ly = m ? (1.0f - ALPHA) * yv.y : 0.0f;
    float rx = fmaf(ALPHA * nn, ax, lx);
    float ry = fmaf(ALPHA * nn, ay, ly);
    rx = fminf(fmaxf(rx, 0.0f), 1.0f);
    ry = fminf(fmaxf(ry, 0.0f), 1.0f);
    float2 r; r.x = rx; r.y = ry;
    ((float2*)(out + ((size_t)node << 6)))[lane] = r;
}

// ------------------------------ launcher -----------------------------------

extern "C" void kernel_launch(void* const* d_in, const int* in_sizes, int n_in,
                              void* d_out, int out_size, void* d_ws, size_t ws_size,
                              hipStream_t stream) {
    (void)in_sizes; (void)n_in; (void)out_size; (void)ws_size;

    const float* y    = (const float*)d_in[0];
    const int*   mask = (const int*)  d_in[1];
    const int*   src  = (const int*)  d_in[2];
    const int*   dst  = (const int*)  d_in[3];
    const float* ew   = (const float*)d_in[4];
    float*       out  = (float*)d_out;

    // Workspace layout (256-B aligned slices, ~52 MB total).
    char* ws = (char*)d_ws;
    size_t off = 0;
    auto take = [&](size_t bytes) -> char* {
        char* p = ws + off;
        off = (off + bytes + 255) & ~(size_t)255;
        return p;
    };
    int*   deg     = (int*)  take(sizeof(int)   * NN);
    int*   row_ptr = (int*)  take(sizeof(int)   * (NN + 1));
    int*   cursor  = (int*)  take(sizeof(int)   * NN);
    float* norm    = (float*)take(sizeof(float) * NN);
    int2*  pairs   = (int2*) take(sizeof(int2)  * NE);
    float* bufA    = (float*)take(sizeof(float) * NN * NC);

    const int T   = 256;
    const int gN  = (NN + T - 1) / T;
    const int gE  = (NE + T - 1) / T;
    const int gV4 = ((NN * NC) / 4 + T - 1) / T;
    const int gW  = (NN + 7) / 8;                      // 8 waves/block, 1 wave/node

    // 1) degrees
    k_zero_i32 <<<gN, T, 0, stream>>>(deg, NN);
    k_count_deg<<<gE, T, 0, stream>>>(dst, deg);
    // 2) row_ptr = exclusive scan(deg)
    k_scan<<<1, 1024, 0, stream>>>(deg, row_ptr);
    // 3) norm + cursor
    k_norm_cursor<<<gN, T, 0, stream>>>(deg, row_ptr, norm, cursor);
    // 4) CSR pairs (src, ew*norm[src])
    k_scatter<<<gE, T, 0, stream>>>(src, dst, ew, norm, cursor, pairs);
    // 5) out0 = mask ? y : 0  (into d_out; 10 layers ping-pong so layer 10
    //    lands back in d_out)
    k_init<<<gV4, T, 0, stream>>>(y, mask, out);
    // 6) 10 propagation layers: d_out -> bufA -> d_out -> ... -> d_out
    const float* rd = out;
    float*       wr = bufA;
    for (int l = 0; l < NLAYERS; ++l) {
        k_layer<<<gW, T, 0, stream>>>(rd, wr, pairs, row_ptr, norm, y, mask);
        const float* t = wr; wr = (float*)rd; rd = t;
    }
}